// GAT_22445499089749
// MI455X (gfx1250) — compile-verified
//
#include <hip/hip_runtime.h>
#include <math.h>

// ---------------------------------------------------------------------------
// GAT layer, faithful to the reference's repeat_interleave quirk:
//   scores[b,i,j] = leaky_relu(A1 * h[b,i,j>>5] + s2[b,i])
//   att = softmax_j(mask(scores));  out = att @ h
// N=1024, C=64, H=32, B=2. All f32; matrix ops via V_WMMA_F32_16X16X4_F32.
// ---------------------------------------------------------------------------

typedef float v2f __attribute__((ext_vector_type(2)));
typedef float v8f __attribute__((ext_vector_type(8)));

#define GAT_N   1024
#define GAT_C   64
#define GAT_H   32
#define GAT_B   2
#define GAT_THR 200000.0f
#define GAT_ALPHA 0.01f
#define GAT_NEG (-1e15f)

// ---------------------------------------------------------------------------
// Kernel 1: h = x @ W^T + b      [B*N, C] x [C, H] -> [B*N, H]
// One wave per 16x16 tile of h. K stepped by 4 with f32 WMMA.
// Fragment layouts (CDNA5 ISA 7.12.2):
//   A 16x4 f32 : lanes 0-15 -> M=lane, VGPR{0,1}=K{0,1}; lanes 16-31 -> K{2,3}
//   B 4x16 f32 : lanes 0-15 -> N=lane, VGPR{0,1}=K{0,1}; lanes 16-31 -> K{2,3}
//   C/D 16x16  : VGPR g -> M=g (lanes 0-15) or M=g+8 (lanes 16-31), N=lane&15
// ---------------------------------------------------------------------------
__global__ __launch_bounds__(32) void gat_proj(const float* __restrict__ x,
                                               const float* __restrict__ w,   // [H, C]
                                               const float* __restrict__ bias, // [H]
                                               float* __restrict__ h) {       // [B*N, H]
  const int tile  = blockIdx.x;          // (B*N/16) * (H/16) tiles
  const int ct    = tile & 1;            // H/16 == 2 column tiles
  const int rt    = tile >> 1;
  const int r0    = rt * 16;
  const int c0    = ct * 16;
  const int lane  = threadIdx.x;
  const int m     = lane & 15;
  const int khalf = (lane >> 4) << 1;    // 0 for lanes 0-15, 2 for lanes 16-31
  const int col   = c0 + m;

  // Fold bias into the accumulator: every element this lane owns has column `col`.
  const float bcol = bias[col];
  v8f acc;
#pragma unroll
  for (int g = 0; g < 8; ++g) acc[g] = bcol;

  const float* __restrict__ xrow = x + (size_t)(r0 + m) * GAT_C;
  const float* __restrict__ wrow = w + (size_t)col * GAT_C;      // B[k][n] = w[n][k]

#pragma unroll
  for (int k = 0; k < GAT_C; k += 4) {
    v2f a, b;
    a.x = xrow[k + khalf + 0];
    a.y = xrow[k + khalf + 1];
    b.x = wrow[k + khalf + 0];
    b.y = wrow[k + khalf + 1];
    acc = __builtin_amdgcn_wmma_f32_16x16x4_f32(false, a, false, b,
                                                (short)0, acc, false, false);
  }

  const int rowbase = r0 + ((lane >> 4) ? 8 : 0);
#pragma unroll
  for (int g = 0; g < 8; ++g)
    h[(size_t)(rowbase + g) * GAT_H + col] = acc[g];
}

// ---------------------------------------------------------------------------
// Kernel 2: masked scores + row softmax -> att [B, N, N]
// One 256-thread block per (b, i) row; each thread owns 4 columns.
// ---------------------------------------------------------------------------
__global__ __launch_bounds__(256) void gat_softmax(const float* __restrict__ h,
                                                   const float* __restrict__ dist,
                                                   const float* __restrict__ a_w, // [2H]
                                                   float* __restrict__ att) {
  const int bi = blockIdx.x;             // b*N + i
  const int b  = bi >> 10;
  const int i  = bi & (GAT_N - 1);
  const int t  = threadIdx.x;

  __shared__ float sh_h[GAT_H];
  __shared__ float sred[256];

  if (t < GAT_H) sh_h[t] = h[(size_t)bi * GAT_H + t];
  __syncthreads();

  // A1 = sum(a_w[0:H]);  s2 = dot(a_w[H:2H], h[b,i,:])  (redundant per thread, cheap)
  float A1 = 0.f, s2 = 0.f;
#pragma unroll
  for (int j = 0; j < GAT_H; ++j) {
    A1 += a_w[j];
    s2 = fmaf(a_w[GAT_H + j], sh_h[j], s2);
  }

  const float* __restrict__ drow = dist + (size_t)i * GAT_N;
  float vals[4];
  float vmax = -INFINITY;
#pragma unroll
  for (int q = 0; q < 4; ++q) {
    const int j = t + 256 * q;
    const float e  = fmaf(A1, sh_h[j >> 5], s2);
    const float sc = (e > 0.f) ? e : GAT_ALPHA * e;       // leaky_relu
    const bool  on = (drow[j] < GAT_THR) || (j == i);     // adj (diag forced 1)
    const float v  = on ? sc : GAT_NEG;
    vals[q] = v;
    vmax = fmaxf(vmax, v);
  }

  // block max
  sred[t] = vmax; __syncthreads();
  for (int s = 128; s > 0; s >>= 1) {
    if (t < s) sred[t] = fmaxf(sred[t], sred[t + s]);
    __syncthreads();
  }
  vmax = sred[0]; __syncthreads();

  float psum = 0.f;
#pragma unroll
  for (int q = 0; q < 4; ++q) {
    const float ev = __expf(vals[q] - vmax);  // exp(NEG - max) underflows to 0 == mask
    vals[q] = ev;
    psum += ev;
  }

  // block sum
  sred[t] = psum; __syncthreads();
  for (int s = 128; s > 0; s >>= 1) {
    if (t < s) sred[t] += sred[t + s];
    __syncthreads();
  }
  const float inv = 1.0f / sred[0];

  float* __restrict__ arow = att + (size_t)bi * GAT_N;
#pragma unroll
  for (int q = 0; q < 4; ++q)
    arow[t + 256 * q] = vals[q] * inv;
}

// ---------------------------------------------------------------------------
// Kernel 3: out = att @ h   per batch: [1024,1024] x [1024,32]
// One wave per 16x16 output tile; K=1024 -> 256 f32 WMMA per wave.
// A rows (att) are contiguous -> the per-lane K pair coalesces into b64 loads.
// ---------------------------------------------------------------------------
__global__ __launch_bounds__(32) void gat_out(const float* __restrict__ att,
                                              const float* __restrict__ h,
                                              float* __restrict__ out) {
  const int tile  = blockIdx.x;          // B * (N/16) * (H/16) = 256
  const int ct    = tile & 1;
  const int rt    = (tile >> 1) & 63;
  const int b     = tile >> 7;
  const int r0    = rt * 16;
  const int c0    = ct * 16;
  const int lane  = threadIdx.x;
  const int m     = lane & 15;
  const int khalf = (lane >> 4) << 1;
  const int col   = c0 + m;

  const float* __restrict__ arow = att + ((size_t)b * GAT_N + r0 + m) * GAT_N + khalf;
  const float* __restrict__ hb   = h + (size_t)b * GAT_N * GAT_H + (size_t)khalf * GAT_H + col;

  v8f acc = {0.f, 0.f, 0.f, 0.f, 0.f, 0.f, 0.f, 0.f};

#pragma unroll 8
  for (int k = 0; k < GAT_N; k += 4) {
    v2f a, bb;
    a.x  = arow[k + 0];
    a.y  = arow[k + 1];
    bb.x = hb[(size_t)k * GAT_H];
    bb.y = hb[(size_t)k * GAT_H + GAT_H];
    acc = __builtin_amdgcn_wmma_f32_16x16x4_f32(false, a, false, bb,
                                                (short)0, acc, false, false);
  }

  const int rowbase = r0 + ((lane >> 4) ? 8 : 0);
  float* __restrict__ ob = out + (size_t)b * GAT_N * GAT_H;
#pragma unroll
  for (int g = 0; g < 8; ++g)
    ob[(size_t)(rowbase + g) * GAT_H + col] = acc[g];
}

// ---------------------------------------------------------------------------
// Launch: inputs in setup_inputs() order: x, dist_mat, proj_w, proj_b, a_w
// Workspace: h [B*N*H] f32 at offset 0, att [B*N*N] f32 after it (~8.25 MB).
// ---------------------------------------------------------------------------
extern "C" void kernel_launch(void* const* d_in, const int* in_sizes, int n_in,
                              void* d_out, int out_size, void* d_ws, size_t ws_size,
                              hipStream_t stream) {
  const float* x      = (const float*)d_in[0];  // [B, N, C]
  const float* dist   = (const float*)d_in[1];  // [N, N]
  const float* proj_w = (const float*)d_in[2];  // [H, C]
  const float* proj_b = (const float*)d_in[3];  // [H]
  const float* a_w    = (const float*)d_in[4];  // [1, 2H]
  float* out = (float*)d_out;                   // [B, N, H]

  float* h   = (float*)d_ws;                            // B*N*H floats
  float* att = (float*)d_ws + (size_t)GAT_B * GAT_N * GAT_H;  // B*N*N floats

  // 1) projection: (B*N/16) * (H/16) = 128 * 2 = 256 waves
  gat_proj<<<256, 32, 0, stream>>>(x, proj_w, proj_b, h);

  // 2) scores + masked softmax: one block per (b, i) row
  gat_softmax<<<GAT_B * GAT_N, 256, 0, stream>>>(h, dist, a_w, att);

  // 3) out = att @ h: B * (N/16) * (H/16) = 256 waves
  gat_out<<<256, 32, 0, stream>>>(att, h, out);
}